// ChiENNMessageKNeighborsSingleDirection_19421842112979
// MI455X (gfx1250) — compile-verified
//
#include <hip/hip_runtime.h>

typedef __attribute__((ext_vector_type(2))) float v2f;
typedef __attribute__((ext_vector_type(4))) float v4f;
typedef __attribute__((ext_vector_type(8))) float v8f;

#define D      128
#define CIRCLE 8
#define PSLOTS 7   // CIRCLE - PAD

// ------- Phase 1 (fused): E1 = x@w1+b1 and E2 = x@w2+b2 in one pass -------
// grid.x = ceil(nrows/16); block = (32,8): 8 wave32s, wave w -> cols [16w,16w+16)
// Both 64KB weight matrices live in LDS (128KB of the 320KB/WGP pool), so each
// A-fragment load from x feeds two v_wmma_f32_16x16x4_f32 ops and x streams
// from HBM exactly once. Tail blocks clamp row0 and recompute a few rows
// (identical values) instead of per-element store guards.
__global__ __launch_bounds__(256) void emb_gemm2_kernel(
    const float* __restrict__ x,
    const float* __restrict__ w1, const float* __restrict__ b1,
    const float* __restrict__ w2, const float* __restrict__ b2,
    float* __restrict__ E1, float* __restrict__ E2, int nrows)
{
    __shared__ float ldsW[2 * D * D];            // [0..16384) = w1, [16384..) = w2
    const int tid = threadIdx.y * 32 + threadIdx.x;
    for (int i = tid; i < (D * D) / 4; i += 256) {
        ((v4f*)ldsW)[i]               = ((const v4f*)w1)[i];
        ((v4f*)ldsW)[i + (D * D) / 4] = ((const v4f*)w2)[i];
    }
    __syncthreads();

    const int lane  = threadIdx.x;
    const int mrow  = lane & 15;              // A-matrix M / D-matrix N index
    const int khalf = (lane < 16) ? 0 : 2;    // A frag: hi half-wave holds K+2..K+3
    const int col0  = threadIdx.y * 16;
    int row0 = blockIdx.x * 16;
    if (row0 > nrows - 16) row0 = nrows - 16; // overlap-recompute tail, no guards

    const float* xrow  = x + (size_t)(row0 + mrow) * D;
    const float* ldsW2 = ldsW + D * D;        // separate base for w2 DS offsets
    const float  bias1 = b1[col0 + mrow];
    const float  bias2 = b2[col0 + mrow];

    v8f c1 = {}, c2 = {};
#pragma unroll
    for (int k0 = 0; k0 < D; k0 += 4) {
        const int kk = k0 + khalf;
        v2f a = *(const v2f*)(xrow + kk);       // one 8B load feeds two WMMAs
        v2f bA, bB;
        bA.x = ldsW[kk * D + col0 + mrow];
        bA.y = ldsW[(kk + 1) * D + col0 + mrow];
        bB.x = ldsW2[kk * D + col0 + mrow];
        bB.y = ldsW2[(kk + 1) * D + col0 + mrow];
        c1 = __builtin_amdgcn_wmma_f32_16x16x4_f32(false, a, false, bA,
                                                   (short)0, c1, false, false);
        c2 = __builtin_amdgcn_wmma_f32_16x16x4_f32(false, a, false, bB,
                                                   (short)0, c2, false, false);
    }

    const int mbase = (lane < 16) ? 0 : 8;      // C/D layout: hi half-wave = M+8
#pragma unroll
    for (int v = 0; v < 8; ++v) {
        const size_t off = (size_t)(row0 + mbase + v) * D + col0 + mrow;
        E1[off] = c1[v] + bias1;
        E2[off] = c2[v] + bias2;
    }
}

// ------- Phase 2+3 fused: out = (gathered msg) @ wf + bf, plus mask -------
// msg rows are gathered from L2-resident E1/E2 on the fly while building A
// fragments; masked rows contribute a zero A-row so they output bf exactly
// as the reference does. Branch-skipping the gather saves ~56% of gather
// bandwidth (average in_degree covers ~44% of slots).
__global__ __launch_bounds__(256) void msg_gemm_kernel(
    const float* __restrict__ E1, const float* __restrict__ E2,
    const float* __restrict__ wf, const float* __restrict__ bf,
    const int* __restrict__ ci,
    float* __restrict__ out_msg, float* __restrict__ out_mask, int nmsg)
{
    __shared__ float ldsW[D * D];
    const int tid = threadIdx.y * 32 + threadIdx.x;
    for (int i = tid; i < (D * D) / 4; i += 256)
        ((v4f*)ldsW)[i] = ((const v4f*)wf)[i];
    __syncthreads();

    const int lane  = threadIdx.x;
    const int mrow  = lane & 15;
    const int khalf = (lane < 16) ? 0 : 2;
    const int col0  = threadIdx.y * 16;
    int row0 = blockIdx.x * 16;
    if (row0 > nmsg - 16) row0 = nmsg - 16;   // overlap-recompute tail, no guards

    const int r = row0 + mrow;
    const int n = r / PSLOTS;
    const int p = r - n * PSLOTS;

    const int* cin = ci + (size_t)n * CIRCLE;
    int msum = 0;
#pragma unroll
    for (int q = 0; q < CIRCLE; ++q) msum += (cin[q] == -1) ? 1 : 0;
    const int  deg  = (msum == CIRCLE) ? 0 : (CIRCLE - 1 - msum);
    const bool mask = (p + 1 <= deg);   // mask==true implies i0,i1 >= 0
    const bool use2 = (deg >= 2);       // second embedding joins only when deg>=2
    const int  i0   = cin[p];
    const int  i1   = cin[p + 1];
    const float* e1row = E1 + (size_t)(i0 < 0 ? 0 : i0) * D;
    const float* e2row = E2 + (size_t)(i1 < 0 ? 0 : i1) * D;
    const float  bias  = bf[col0 + mrow];

    v8f c = {};
#pragma unroll
    for (int k0 = 0; k0 < D; k0 += 4) {
        const int kk = k0 + khalf;
        v2f a; a.x = 0.0f; a.y = 0.0f;
        if (mask) {                      // divergent gather, reconverges pre-WMMA
            a = *(const v2f*)(e1row + kk);
            if (use2) {
                v2f a2 = *(const v2f*)(e2row + kk);
                a.x += a2.x; a.y += a2.y;
            }
        }
        v2f bb;
        bb.x = ldsW[kk * D + col0 + mrow];
        bb.y = ldsW[(kk + 1) * D + col0 + mrow];
        c = __builtin_amdgcn_wmma_f32_16x16x4_f32(false, a, false, bb,
                                                  (short)0, c, false, false);
    }

    const int mbase = (lane < 16) ? 0 : 8;
#pragma unroll
    for (int v = 0; v < 8; ++v)   // NT stores: keep 358MB result out of L2
        __builtin_nontemporal_store(c[v] + bias,
                                    &out_msg[(size_t)(row0 + mbase + v) * D
                                             + col0 + mrow]);
    if (threadIdx.y == 0 && lane < 16)
        __builtin_nontemporal_store(mask ? 1.0f : 0.0f, out_mask + r);
}

extern "C" void kernel_launch(void* const* d_in, const int* in_sizes, int n_in,
                              void* d_out, int out_size, void* d_ws, size_t ws_size,
                              hipStream_t stream)
{
    const float* x  = (const float*)d_in[0];
    const float* w1 = (const float*)d_in[1];
    const float* b1 = (const float*)d_in[2];
    const float* w2 = (const float*)d_in[3];
    const float* b2 = (const float*)d_in[4];
    const float* wf = (const float*)d_in[5];
    const float* bf = (const float*)d_in[6];
    const int*   ci = (const int*)d_in[7];

    const int N = in_sizes[0] / D;          // 100000
    float* E1 = (float*)d_ws;               // N*128 f32
    float* E2 = E1 + (size_t)N * D;         // N*128 f32 (102.4 MB total, L2-resident)
    float* out_msg  = (float*)d_out;
    float* out_mask = out_msg + (size_t)N * PSLOTS * D;

    dim3 blk(32, 8, 1);
    emb_gemm2_kernel<<<(N + 15) / 16, blk, 0, stream>>>(x, w1, b1, w2, b2,
                                                        E1, E2, N);

    const int nmsg = N * PSLOTS;            // 700000
    msg_gemm_kernel<<<(nmsg + 15) / 16, blk, 0, stream>>>(E1, E2, wf, bf, ci,
                                                          out_msg, out_mask, nmsg);
}